// PillarNet_10247791969011
// MI455X (gfx1250) — compile-verified
//
#include <hip/hip_runtime.h>
#include <hip/hip_bf16.h>

// ---------------- problem constants (from reference) ----------------
#define N_POINTS 1200000
#define GXC 512
#define GYC 512
#define BATCHC 4
#define NUM_SEG (BATCHC * GXC * GYC)   // 1,048,576 dense voxel ids
#define NBINS (NUM_SEG + 1)            // +1 trash bin
#define TPB 256                        // 8 wave32 waves per block

#define AS1 __attribute__((address_space(1)))
#define AS3 __attribute__((address_space(3)))

// CDNA5 async global<->LDS paths (gfx1250), ASYNCcnt-tracked.
// Load builtin confirmed present: (AS1 int* src, AS3 int* dst, imm off, imm cpol).
#if defined(__gfx1250__) && __has_builtin(__builtin_amdgcn_global_load_async_to_lds_b32)
#define HAVE_ASYNC_LD 1
#else
#define HAVE_ASYNC_LD 0
#endif
#if defined(__gfx1250__) && __has_builtin(__builtin_amdgcn_global_store_async_from_lds_b32)
#define HAVE_ASYNC_ST 1
#else
#define HAVE_ASYNC_ST 0
#endif

__device__ __forceinline__ void wait_async_zero() {
#if defined(__gfx1250__)
#if __has_builtin(__builtin_amdgcn_s_wait_asynccnt)
    __builtin_amdgcn_s_wait_asynccnt(0);
#else
    asm volatile("s_wait_asynccnt 0" ::: "memory");
#endif
    asm volatile("" ::: "memory");   // keep LDS reads after the wait
#endif
}

__device__ __forceinline__ void wait_ds_zero() {
#if defined(__gfx1250__)
    asm volatile("s_wait_dscnt 0x0" ::: "memory");  // drain LDS writes + compiler barrier
#endif
}

// Shared helper: voxel coordinates exactly as the reference computes them.
//   pc = (p - pr[:3]) / vs ;  mask on x/y ;  coords = trunc-to-int
__device__ __forceinline__ void voxelize(float px, float py,
                                         int& ix, int& iy, bool& mask) {
    float cx = (px - (-51.2f)) / 0.2f;
    float cy = (py - (-51.2f)) / 0.2f;
    mask = (cx >= 0.0f) && (cx < (float)GXC) && (cy >= 0.0f) && (cy < (float)GYC);
    ix = (int)cx;   // trunc == floor for the in-range (>=0) values that matter
    iy = (int)cy;
}

// ---------------- kernel 1: zero the bin array (sum_x,sum_y,sum_z,count) ----
__global__ void pillar_init_bins(float4* __restrict__ bins) {
    int i = blockIdx.x * TPB + threadIdx.x;
    if (i < NBINS) bins[i] = make_float4(0.f, 0.f, 0.f, 0.f);
}

// ---------------- kernel 2: scatter-add xyz + count, emit seg ---------------
__global__ void pillar_accumulate(const float* __restrict__ pts,
                                  float* __restrict__ bins,
                                  float* __restrict__ seg_out) {
    int i = blockIdx.x * TPB + threadIdx.x;
    if (i >= N_POINTS) return;
    const float* p = pts + (size_t)i * 5;
    float pb = p[0], px = p[1], py = p[2], pz = p[3];
    int ix, iy; bool mask;
    voxelize(px, py, ix, iy, mask);
    int b = (int)pb;
    int seg = mask ? (b * (GXC * GYC) + ix * GYC + iy) : NUM_SEG;
    seg_out[i] = (float)seg;                     // flat-float output convention
    if (mask) {
        float* bin = bins + (size_t)seg * 4;     // ~1.14 pts/bin: contention-free
        atomicAdd(bin + 0, px);                  // global_atomic_add_f32
        atomicAdd(bin + 1, py);
        atomicAdd(bin + 2, pz);
        atomicAdd(bin + 3, 1.0f);
    }
}

// ---------------- kernel 3: sums -> means in place; write grid_size ---------
__global__ void pillar_finalize_means(float4* __restrict__ bins,
                                      float* __restrict__ grid_out) {
    int i = blockIdx.x * TPB + threadIdx.x;
    if (i == 0) { grid_out[0] = (float)GYC; grid_out[1] = (float)GXC; }
    if (i >= NBINS) return;
    float4 v = bins[i];
    float c = fmaxf(v.w, 1.0f);
    bins[i] = make_float4(v.x / c, v.y / c, v.z / c, v.w);
}

// ---------------- kernel 4: gather means (L2-resident), emit 9 features -----
// Per-wave LDS transpose so the 43 MB feature stream is written as fully
// coalesced 128B stores (each wave owns a 288-float slot; N_POINTS % 32 == 0,
// so waves are either fully valid or fully returned -> no block barrier).
__global__ void pillar_features(const float* __restrict__ pts,
                                const float4* __restrict__ bins,
                                float* __restrict__ feat_out) {
    __shared__ float lds_feat[TPB * 9];          // transposed store staging
    int i = blockIdx.x * TPB + threadIdx.x;
    int lane = threadIdx.x & 31;
    int wave = threadIdx.x >> 5;
    float p0, p1, p2, p3, p4;
#if HAVE_ASYNC_LD
    __shared__ float lds_pts[TPB * 5];           // async-in staging
    if (i < N_POINTS) {
        AS1 int* gp = (AS1 int*)const_cast<float*>(pts + (size_t)i * 5);
        AS3 int* lp = (AS3 int*)&lds_pts[threadIdx.x * 5];
        __builtin_amdgcn_global_load_async_to_lds_b32(gp, lp, 0, 0);
        __builtin_amdgcn_global_load_async_to_lds_b32(gp, lp, 4, 0);
        __builtin_amdgcn_global_load_async_to_lds_b32(gp, lp, 8, 0);
        __builtin_amdgcn_global_load_async_to_lds_b32(gp, lp, 12, 0);
        __builtin_amdgcn_global_load_async_to_lds_b32(gp, lp, 16, 0);
    }
    wait_async_zero();          // only this wave's own slots are read: no barrier
    if (i >= N_POINTS) return;
    p0 = lds_pts[threadIdx.x * 5 + 0];
    p1 = lds_pts[threadIdx.x * 5 + 1];
    p2 = lds_pts[threadIdx.x * 5 + 2];
    p3 = lds_pts[threadIdx.x * 5 + 3];
    p4 = lds_pts[threadIdx.x * 5 + 4];
#else
    if (i >= N_POINTS) return;
    const float* p = pts + (size_t)i * 5;
    p0 = p[0]; p1 = p[1]; p2 = p[2]; p3 = p[3]; p4 = p[4];
#endif
    int ix, iy; bool mask;
    voxelize(p1, p2, ix, iy, mask);
    int b = (int)p0;
    int seg = mask ? (b * (GXC * GYC) + ix * GYC + iy) : NUM_SEG;
    float4 m = bins[seg];                        // 16.8 MB table: L2-resident
    float mf = mask ? 1.0f : 0.0f;
    float ctrx = (float)ix * 0.2f + 0.1f + (-51.2f);
    float ctry = (float)iy * 0.2f + 0.1f + (-51.2f);
    float o[9];
    o[0] = p1; o[1] = p2; o[2] = p3; o[3] = p4;  // points[:,1:]
    o[4] = p1 - m.x; o[5] = p2 - m.y; o[6] = p3 - m.z;   // f_cluster
    o[7] = p1 - ctrx; o[8] = p2 - ctry;                  // f_center

    // Scatter row into this wave's LDS slot (stride 9 is coprime with the
    // bank count -> conflict-free), then emit 9 coalesced 128B stores.
    float* slot = &lds_feat[wave * 32 * 9];
#pragma unroll
    for (int c = 0; c < 9; ++c) slot[lane * 9 + c] = o[c] * mf;
    size_t base = (size_t)blockIdx.x * (TPB * 9) + (size_t)wave * (32 * 9);
#if HAVE_ASYNC_ST
    wait_ds_zero();                              // LDS writes visible to async pipe
    AS1 int* gdst = (AS1 int*)(feat_out + base + lane);
    AS3 int* lsrc = (AS3 int*)&slot[lane];
    __builtin_amdgcn_global_store_async_from_lds_b32(gdst, lsrc, 0 * 128, 0);
    __builtin_amdgcn_global_store_async_from_lds_b32(gdst, lsrc, 1 * 128, 0);
    __builtin_amdgcn_global_store_async_from_lds_b32(gdst, lsrc, 2 * 128, 0);
    __builtin_amdgcn_global_store_async_from_lds_b32(gdst, lsrc, 3 * 128, 0);
    __builtin_amdgcn_global_store_async_from_lds_b32(gdst, lsrc, 4 * 128, 0);
    __builtin_amdgcn_global_store_async_from_lds_b32(gdst, lsrc, 5 * 128, 0);
    __builtin_amdgcn_global_store_async_from_lds_b32(gdst, lsrc, 6 * 128, 0);
    __builtin_amdgcn_global_store_async_from_lds_b32(gdst, lsrc, 7 * 128, 0);
    __builtin_amdgcn_global_store_async_from_lds_b32(gdst, lsrc, 8 * 128, 0);
    // S_ENDPGM performs an implicit wait-idle: ASYNCcnt drains before exit.
#else
#pragma unroll
    for (int k = 0; k < 9; ++k)
        feat_out[base + (size_t)k * 32 + lane] = slot[k * 32 + lane];
#endif
}

// ---------------- launch ----------------------------------------------------
extern "C" void kernel_launch(void* const* d_in, const int* in_sizes, int n_in,
                              void* d_out, int out_size, void* d_ws, size_t ws_size,
                              hipStream_t stream) {
    (void)in_sizes; (void)n_in; (void)out_size; (void)ws_size;
    const float* pts = (const float*)d_in[0];
    float* out  = (float*)d_out;
    float* feat = out;                                   // N*9
    float* seg  = out + (size_t)N_POINTS * 9;            // N
    float* grid = seg + N_POINTS;                        // 2
    float* bins = (float*)d_ws;                          // NBINS * float4 = ~16.8 MB

    const int binBlocks = (NBINS + TPB - 1) / TPB;
    const int ptBlocks  = (N_POINTS + TPB - 1) / TPB;

    pillar_init_bins<<<binBlocks, TPB, 0, stream>>>((float4*)bins);
    pillar_accumulate<<<ptBlocks, TPB, 0, stream>>>(pts, bins, seg);
    pillar_finalize_means<<<binBlocks, TPB, 0, stream>>>((float4*)bins, grid);
    pillar_features<<<ptBlocks, TPB, 0, stream>>>(pts, (const float4*)bins, feat);
}